// DistanceEncoder_35107062677895
// MI455X (gfx1250) — compile-verified
//
#include <hip/hip_runtime.h>

#define HH   16
#define ATOM 4
#define NB   2
#define R    384
#define NN   (R * ATOM)   // 1536
#define RR   (R * R)      // 147456

typedef __attribute__((ext_vector_type(2))) float v2f;
typedef __attribute__((ext_vector_type(8))) float v8f;

__device__ __forceinline__ float4 f4add(float4 a, float4 b) {
    return make_float4(a.x + b.x, a.y + b.y, a.z + b.z, a.w + b.w);
}
__device__ __forceinline__ float4 f4fma(float4 a, float4 b, float4 c) {
    return make_float4(fmaf(a.x, b.x, c.x), fmaf(a.y, b.y, c.y),
                       fmaf(a.z, b.z, c.z), fmaf(a.w, b.w, c.w));
}

__device__ __forceinline__ int dir_bin(float r) {
    // dl = [-1e8, linspace(-50,50,127)], du = [dl[1:], 1e8]; argmax(mask)+1
    if (r <= -50.0f) return 1;
    if (r >=  50.0f) return 128;
    int j = 1 + (int)((r + 50.0f) * (126.0f / 100.0f));
    if (j > 127) j = 127;
    return j + 1;
}

// -------- Kernel A: coord_update + pos_emb -> d_out (B,H,R,R layout) --------
__global__ void spatial_pos_kernel(const float* __restrict__ coords,
                                   const float* __restrict__ rpe_table,
                                   const float* __restrict__ relpos_w,
                                   const float* __restrict__ relpos_b,
                                   float* __restrict__ out) {
    int gid = blockIdx.x * blockDim.x + threadIdx.x;
    if (gid >= NB * RR) return;
    int b   = gid / RR;
    int rem = gid - b * RR;
    int p   = rem / R;
    int q   = rem - p * R;

    const float* cb  = coords + (size_t)b * NN * 3;
    const float* nq  = cb + (size_t)(4 * q + 0) * 3;  // n_at[q]
    const float* caq = cb + (size_t)(4 * q + 1) * 3;  // ca[q]
    const float* ctq = cb + (size_t)(4 * q + 2) * 3;  // c_at[q]
    const float* cap = cb + (size_t)(4 * p + 1) * 3;  // ca[p]

    float cax = caq[0], cay = caq[1], caz = caq[2];
    float dx = cap[0] - cax, dy = cap[1] - cay, dz = cap[2] - caz;

    // distance bin (diff + 1e-6 before squaring, per reference)
    float ex = dx + 1e-6f, ey = dy + 1e-6f, ez = dz + 1e-6f;
    float dist = sqrtf(ex * ex + ey * ey + ez * ez);
    int d_idx;
    if (dist <= 3.0f) d_idx = 1;
    else {
        int k = (int)((dist - 3.0f) * (127.0f / 77.0f));
        if (k > 127) k = 127;
        d_idx = k + 1;
    }

    // frame axes at q
    float xx = nq[0] - cax, xy = nq[1] - cay, xz = nq[2] - caz;
    float xn = fmaxf(sqrtf(xx * xx + xy * xy + xz * xz), 1e-12f);
    xx /= xn; xy /= xn; xz /= xn;
    float tx = ctq[0] - cax, ty = ctq[1] - cay, tz = ctq[2] - caz;
    float zx = xy * tz - xz * ty;
    float zy = xz * tx - xx * tz;
    float zz = xx * ty - xy * tx;
    float zn = fmaxf(sqrtf(zx * zx + zy * zy + zz * zz), 1e-12f);
    zx /= zn; zy /= zn; zz /= zn;
    float yx = zy * xz - zz * xy;
    float yy = zz * xx - zx * xz;
    float yz = zx * xy - zy * xx;

    float rx = dx * xx + dy * xy + dz * xz;
    float ry = dx * yx + dy * yy + dz * yz;
    float rz = dx * zx + dy * zy + dz * zz;
    int gx = dir_bin(rx), gy = dir_bin(ry), gz = dir_bin(rz);

    int ridx = p - q;
    ridx = (ridx < -32 ? -32 : (ridx > 32 ? 32 : ridx)) + 32;

    const float4* w4  = (const float4*)relpos_w;
    const float4* rp4 = (const float4*)rpe_table;
    const float4* b4  = (const float4*)relpos_b;

    size_t obase = (size_t)b * HH * RR + (size_t)p * R + q;
#pragma unroll
    for (int j = 0; j < 4; ++j) {
        float4 a = w4[(size_t)d_idx * 4 + j];
        a = f4add(a, w4[(size_t)(129 + gx) * 4 + j]);
        a = f4add(a, w4[(size_t)(258 + gy) * 4 + j]);
        a = f4add(a, w4[(size_t)(387 + gz) * 4 + j]);
        a = f4add(a, rp4[(size_t)ridx * 4 + j]);
        a = f4add(a, b4[j]);
        out[obase + (size_t)(4 * j + 0) * RR] = a.x;
        out[obase + (size_t)(4 * j + 1) * RR] = a.y;
        out[obase + (size_t)(4 * j + 2) * RR] = a.z;
        out[obase + (size_t)(4 * j + 3) * RR] = a.w;
    }
}

// -------- Kernel B: 6-table gather + param + 4x4 pool -> g in d_ws --------
__global__ void gbf_gather_kernel(const float* __restrict__ dist,
                                  const float* __restrict__ e1, const float* __restrict__ e2,
                                  const float* __restrict__ e3, const float* __restrict__ e4,
                                  const float* __restrict__ e5, const float* __restrict__ e6,
                                  const float* __restrict__ param,
                                  float* __restrict__ g) {
    int gid = blockIdx.x * blockDim.x + threadIdx.x;
    if (gid >= NB * RR) return;
    int b   = gid / RR;
    int rem = gid - b * RR;
    int p   = rem / R;
    int q   = rem - p * R;

    const float4* E1 = (const float4*)e1;
    const float4* E2 = (const float4*)e2;
    const float4* E3 = (const float4*)e3;
    const float4* E4 = (const float4*)e4;
    const float4* E5 = (const float4*)e5;
    const float4* E6 = (const float4*)e6;
    const float4* P4 = (const float4*)param;

    float4 acc[4];
    acc[0] = acc[1] = acc[2] = acc[3] = make_float4(0.f, 0.f, 0.f, 0.f);

    const float* drow = dist + (size_t)b * NN * NN + (size_t)(4 * p) * NN + 4 * q;
#pragma unroll
    for (int a1 = 0; a1 < 4; ++a1) {
        float4 d4 = *(const float4*)(drow + (size_t)a1 * NN);
        float dv[4] = {d4.x, d4.y, d4.z, d4.w};
#pragma unroll
        for (int a2 = 0; a2 < 4; ++a2) {
            float d = dv[a2] > 126.0f ? 127.0f : dv[a2];
            // idx6 = floor(d*128); power-of-two scalings are exact in fp32,
            // so the per-table indices are exactly idx6 >> {0,2,4,6,8,10}.
            int i6 = (int)(d * 128.0f);
            int i5 = i6 >> 2, i4 = i6 >> 4, i3 = i6 >> 6, i2 = i6 >> 8, i1 = i6 >> 10;
            int pb = (a1 * 4 + a2) * 4;
#pragma unroll
            for (int j = 0; j < 4; ++j) {
                float4 s = E6[(size_t)i6 * 4 + j];
                s = f4add(s, E5[(size_t)i5 * 4 + j]);
                s = f4add(s, E4[(size_t)i4 * 4 + j]);
                s = f4add(s, E3[(size_t)i3 * 4 + j]);
                s = f4add(s, E2[(size_t)i2 * 4 + j]);
                s = f4add(s, E1[(size_t)i1 * 4 + j]);
                acc[j] = f4fma(s, P4[pb + j], acc[j]);
            }
        }
    }
    float4* gout = (float4*)g;
#pragma unroll
    for (int j = 0; j < 4; ++j) gout[(size_t)gid * 4 + j] = acc[j];
}

// -------- Kernel C: h = gelu(g@w1+b1)@w2+b2 via WMMA; out += h --------
__global__ void mlp_wmma_kernel(const float* __restrict__ g,
                                const float* __restrict__ w1, const float* __restrict__ b1,
                                const float* __restrict__ w2, const float* __restrict__ b2,
                                float* __restrict__ out) {
    __shared__ float lds[8 * 256];  // 1 KB per wave, 8 waves
    int wave = threadIdx.x >> 5;
    int lane = threadIdx.x & 31;
    int tile = blockIdx.x * 8 + wave;           // 18432 tiles total, grid exact
    int m_base = tile * 16;
    int b   = m_base / RR;
    int rem = m_base - b * RR;
    int p   = rem / R;
    int q0  = rem - p * R;                       // multiple of 16 (R % 16 == 0)
    int n   = lane & 15;                         // = A-row m and B/D column
    int hi  = lane >> 4;

    // A (g tile): lane holds A[m=n][k0+2*hi], A[m=n][k0+2*hi+1] per chunk
    const float* grow = g + (size_t)(m_base + n) * 16;
    v2f a0 = *(const v2f*)(grow + 0  + 2 * hi);
    v2f a1 = *(const v2f*)(grow + 4  + 2 * hi);
    v2f a2 = *(const v2f*)(grow + 8  + 2 * hi);
    v2f a3 = *(const v2f*)(grow + 12 + 2 * hi);

    // B (w1, row-major K x N): lane holds B[k0+2*hi][n], B[k0+2*hi+1][n]
#define LOADW(dst, W, k0)                                   \
    { dst.x = (W)[((k0) + 2 * hi) * 16 + n];                \
      dst.y = (W)[((k0) + 2 * hi + 1) * 16 + n]; }
    v2f bw0, bw1, bw2, bw3;
    LOADW(bw0, w1, 0)  LOADW(bw1, w1, 4)  LOADW(bw2, w1, 8)  LOADW(bw3, w1, 12)

    v8f c = {0.f, 0.f, 0.f, 0.f, 0.f, 0.f, 0.f, 0.f};
    c = __builtin_amdgcn_wmma_f32_16x16x4_f32(false, a0, false, bw0, (short)0, c, false, false);
    c = __builtin_amdgcn_wmma_f32_16x16x4_f32(false, a1, false, bw1, (short)0, c, false, false);
    c = __builtin_amdgcn_wmma_f32_16x16x4_f32(false, a2, false, bw2, (short)0, c, false, false);
    c = __builtin_amdgcn_wmma_f32_16x16x4_f32(false, a3, false, bw3, (short)0, c, false, false);

    // bias + exact GELU, stage to LDS in row-major [m][n] (D: M = i + 8*hi, N = n)
    float bias1 = b1[n];
    float* L = lds + wave * 256;
#pragma unroll
    for (int i = 0; i < 8; ++i) {
        float x = c[i] + bias1;
        L[(i + 8 * hi) * 16 + n] = 0.5f * x * (1.0f + erff(x * 0.70710678118654752f));
    }
    __syncthreads();  // conservative; hazard is intra-wave store->load via LDS

    v2f h0 = *(const v2f*)(L + n * 16 + 0  + 2 * hi);
    v2f h1 = *(const v2f*)(L + n * 16 + 4  + 2 * hi);
    v2f h2 = *(const v2f*)(L + n * 16 + 8  + 2 * hi);
    v2f h3 = *(const v2f*)(L + n * 16 + 12 + 2 * hi);

    v2f cw0, cw1, cw2, cw3;
    LOADW(cw0, w2, 0)  LOADW(cw1, w2, 4)  LOADW(cw2, w2, 8)  LOADW(cw3, w2, 12)
#undef LOADW

    v8f c2 = {0.f, 0.f, 0.f, 0.f, 0.f, 0.f, 0.f, 0.f};
    c2 = __builtin_amdgcn_wmma_f32_16x16x4_f32(false, h0, false, cw0, (short)0, c2, false, false);
    c2 = __builtin_amdgcn_wmma_f32_16x16x4_f32(false, h1, false, cw1, (short)0, c2, false, false);
    c2 = __builtin_amdgcn_wmma_f32_16x16x4_f32(false, h2, false, cw2, (short)0, c2, false, false);
    c2 = __builtin_amdgcn_wmma_f32_16x16x4_f32(false, h3, false, cw3, (short)0, c2, false, false);

    // out[(b*16+n), p, q0 + 8*hi + i] += c2[i] + b2[n]; 8 consecutive q per lane
    float bias2 = b2[n];
    size_t obase = (size_t)(b * 16 + n) * RR + (size_t)p * R + q0 + 8 * hi;
    float4 o0 = *(float4*)(out + obase);
    o0.x += c2[0] + bias2;  o0.y += c2[1] + bias2;
    o0.z += c2[2] + bias2;  o0.w += c2[3] + bias2;
    *(float4*)(out + obase) = o0;
    float4 o1 = *(float4*)(out + obase + 4);
    o1.x += c2[4] + bias2;  o1.y += c2[5] + bias2;
    o1.z += c2[6] + bias2;  o1.w += c2[7] + bias2;
    *(float4*)(out + obase + 4) = o1;
}

extern "C" void kernel_launch(void* const* d_in, const int* in_sizes, int n_in,
                              void* d_out, int out_size, void* d_ws, size_t ws_size,
                              hipStream_t stream) {
    const float* coords = (const float*)d_in[0];
    const float* dist   = (const float*)d_in[1];
    const float* e1     = (const float*)d_in[2];
    const float* e2     = (const float*)d_in[3];
    const float* e3     = (const float*)d_in[4];
    const float* e4     = (const float*)d_in[5];
    const float* e5     = (const float*)d_in[6];
    const float* e6     = (const float*)d_in[7];
    const float* param  = (const float*)d_in[8];
    const float* w1     = (const float*)d_in[9];
    const float* b1     = (const float*)d_in[10];
    const float* w2     = (const float*)d_in[11];
    const float* b2     = (const float*)d_in[12];
    const float* rpe    = (const float*)d_in[13];
    const float* rw     = (const float*)d_in[14];
    const float* rb     = (const float*)d_in[15];
    float* out = (float*)d_out;
    float* g   = (float*)d_ws;  // 294912 * 16 floats = 18.9 MB

    const int total = NB * RR;                        // 294912
    spatial_pos_kernel<<<total / 256, 256, 0, stream>>>(coords, rpe, rw, rb, out);
    gbf_gather_kernel<<<total / 256, 256, 0, stream>>>(dist, e1, e2, e3, e4, e5, e6, param, g);
    mlp_wmma_kernel<<<(total / 16) / 8, 256, 0, stream>>>(g, w1, b1, w2, b2, out);
}